// PAAPostProcessor_89154931130508
// MI455X (gfx1250) — compile-verified
//
#include <hip/hip_runtime.h>
#include <stdint.h>

#define BBOX_XFORM_CLIP 4.135166556742356f
#define NB   8
#define TT   32
#define HW   16384
#define WW   128
#define IMG_MAX 1023.0f

// ---- output layout (flat, return order) ----
// matched      : [0      , 1024)   (8,32,4)
// matched_iou  : [1024   , 1280)   (8,32)
// peak_cls     : [1280   , 9472)   (8192)
// peak_iou     : [9472   , 17664)  (8192)
// peak_det     : [17664  , 50432)  (8192,4)
// max_iou_mean : [50432  , 50440)  (8)

// ---- CDNA5 async global->LDS staging (ASYNCcnt pipeline) ----
__device__ __forceinline__ void async_load_b128(uint32_t lds_off, const void* gaddr) {
    asm volatile("global_load_async_to_lds_b128 %0, %1, off"
                 :: "v"(lds_off), "v"((uint64_t)(uintptr_t)gaddr)
                 : "memory");
}
#define WAIT_ASYNCCNT(n) asm volatile("s_wait_asynccnt " #n ::: "memory")

__device__ __forceinline__ float4 decode_box(float dx, float dy, float dw, float dh, float4 a) {
    float w  = a.z - a.x + 1.0f;
    float h  = a.w - a.y + 1.0f;
    float cx = a.x + 0.5f * w;
    float cy = a.y + 0.5f * h;
    dx *= 0.1f;                              // /wx
    dy *= 0.1f;                              // /wy
    dw = fminf(dw * 0.2f, BBOX_XFORM_CLIP);  // /ww, clip
    dh = fminf(dh * 0.2f, BBOX_XFORM_CLIP);  // /wh, clip
    float pcx = dx * w + cx;
    float pcy = dy * h + cy;
    float pw  = __expf(dw) * w;
    float ph  = __expf(dh) * h;
    return make_float4(pcx - 0.5f * pw, pcy - 0.5f * ph,
                       pcx + 0.5f * pw - 1.0f, pcy + 0.5f * ph - 1.0f);
}

// ---------------- kernel 1: decode all anchors -> d_ws ----------------
__global__ __launch_bounds__(256) void decode_whole_kernel(
    const float* __restrict__ reg,        // (N,4,HW)
    const float4* __restrict__ anchors,   // (N,HW,4)
    float4* __restrict__ whole)           // (N,HW,4)
{
    int idx = blockIdx.x * 256 + threadIdx.x;   // 0 .. N*HW-1 (exact)
    int n = idx >> 14;
    int j = idx & (HW - 1);
    const float* r = reg + n * 4 * HW + j;
    float dx = r[0];
    float dy = r[HW];
    float dw = r[2 * HW];
    float dh = r[3 * HW];
    whole[idx] = decode_box(dx, dy, dw, dh, anchors[idx]);
}

// ------------- kernel 2: per-(n,t) IoU max/argmax over HW -------------
__global__ __launch_bounds__(256) void match_kernel(
    const float4* __restrict__ whole,     // (N,HW,4) from d_ws
    const float4* __restrict__ tboxes,    // (N,T,4)
    float* __restrict__ out_matched,      // (N,T,4)
    float* __restrict__ out_miou)         // (N,T)
{
    __shared__ float4 tile[2][256];       // async double buffer
    __shared__ float  s_iou[256];
    __shared__ int    s_idx[256];
    __shared__ float4 s_box[256];

    const int tid = threadIdx.x;
    const int n   = blockIdx.x >> 5;
    const int t   = blockIdx.x & 31;

    const float4 tb = tboxes[n * TT + t];
    const float  ta = (tb.z - tb.x + 1.0f) * (tb.w - tb.y + 1.0f);

    const float4* src = whole + n * HW;

    // prime buffer 0
    async_load_b128((uint32_t)(uintptr_t)&tile[0][tid], src + tid);

    float  best = -1.0f;
    int    bidx = 0x7FFFFFFF;
    float4 bb   = make_float4(0.f, 0.f, 0.f, 0.f);

    const int NTILES = HW / 256;          // 64
    for (int it = 0; it < NTILES; ++it) {
        const int buf = it & 1;
        if (it + 1 < NTILES) {
            // issue next tile, then wait until only it is outstanding
            async_load_b128((uint32_t)(uintptr_t)&tile[buf ^ 1][tid],
                            src + (it + 1) * 256 + tid);
            WAIT_ASYNCCNT(1);
        } else {
            WAIT_ASYNCCNT(0);
        }
        // each lane consumes exactly the LDS slot its own async op filled
        float4 b = tile[buf][tid];

        float ltx = fmaxf(tb.x, b.x);
        float lty = fmaxf(tb.y, b.y);
        float rbx = fminf(tb.z, b.z);
        float rby = fminf(tb.w, b.w);
        float iw  = fmaxf(rbx - ltx + 1.0f, 0.0f);
        float ih  = fmaxf(rby - lty + 1.0f, 0.0f);
        float inter = iw * ih;
        float ab  = (b.z - b.x + 1.0f) * (b.w - b.y + 1.0f);
        float iou = inter / (ta + ab - inter);

        int j = it * 256 + tid;           // strictly increasing per thread:
        if (iou > best) {                 // '>' keeps first occurrence
            best = iou; bidx = j; bb = b;
        }
    }

    s_iou[tid] = best;
    s_idx[tid] = bidx;
    s_box[tid] = bb;
    __syncthreads();

    // tree reduction; tie-break toward lower index (argmax first-occurrence)
    for (int s = 128; s > 0; s >>= 1) {
        if (tid < s) {
            float o  = s_iou[tid + s];
            int   oi = s_idx[tid + s];
            float c  = s_iou[tid];
            int   ci = s_idx[tid];
            if (o > c || (o == c && oi < ci)) {
                s_iou[tid] = o;
                s_idx[tid] = oi;
                s_box[tid] = s_box[tid + s];
            }
        }
        __syncthreads();
    }

    if (tid == 0) {
        float4 m = s_box[0];
        float x1 = fminf(fmaxf(m.x, 0.0f), IMG_MAX);
        float y1 = fminf(fmaxf(m.y, 0.0f), IMG_MAX);
        float x2 = fminf(fmaxf(m.z, 0.0f), IMG_MAX);
        float y2 = fminf(fmaxf(m.w, 0.0f), IMG_MAX);
        int o = (n * TT + t) * 4;
        out_matched[o + 0] = x1;
        out_matched[o + 1] = y1;
        out_matched[o + 2] = x2;
        out_matched[o + 3] = y2;
        float ws = x2 - x1 + 1.0f;
        float hs = y2 - y1 + 1.0f;
        bool keep = (ws >= 0.0f) && (hs >= 0.0f);   // MIN_SIZE == 0
        out_miou[n * TT + t] = keep ? s_iou[0] : 0.0f;
    }
}

// ---------------- kernel 3: peak gathers + peak decode ----------------
__global__ __launch_bounds__(256) void peaks_kernel(
    const float* __restrict__ box_cls,    // (N,80,H,W)
    const float* __restrict__ reg,        // (N,4,HW)
    const float* __restrict__ iou_pred,   // (N,1,H,W)
    const int4*  __restrict__ pcls,       // (P,4)  b,c,y,x
    const int4*  __restrict__ preg,       // (P,4)  b,c,y,x
    const float* __restrict__ logit,      // (P)
    const float4* __restrict__ anchors,   // (N,HW,4)
    float* __restrict__ out)
{
    int i = blockIdx.x * 256 + threadIdx.x;   // 0..8191 exact
    int4 c  = pcls[i];
    int4 rg = preg[i];
    float l = logit[i];
    float p = 1.0f / (1.0f + __expf(-l));
    bool positive = p > 0.05f;

    float pcl = 0.0f, pio = 0.0f;
    if (positive) {
        int ci = ((c.x * 80 + c.y) * WW + c.z) * WW + c.w;
        int ri = ((rg.x * 1 + rg.y) * WW + rg.z) * WW + rg.w;
        pcl = 1.0f / (1.0f + __expf(-box_cls[ci]));
        pio = 1.0f / (1.0f + __expf(-iou_pred[ri]));
    }
    out[1280 + i] = pcl;
    out[9472 + i] = pio;

    int base = rg.x * 4 * HW + rg.z * WW + rg.w;
    float dx = reg[base];
    float dy = reg[base + HW];
    float dw = reg[base + 2 * HW];
    float dh = reg[base + 3 * HW];
    float4 a = anchors[rg.x * HW + rg.z * WW + rg.w];
    float4 d = decode_box(dx, dy, dw, dh, a);
    float sc = positive ? 1.0f : 0.0f;
    ((float4*)(out + 17664))[i] =
        make_float4(d.x * sc, d.y * sc, d.z * sc, d.w * sc);
}

// ------------- kernel 4: deterministic per-batch mean -------------
__global__ void mean_kernel(float* __restrict__ out) {
    int tid = threadIdx.x;
    if (tid < NB) {
        float s = 0.0f;
        for (int t = 0; t < TT; ++t) s += out[1024 + tid * TT + t];
        out[50432 + tid] = s * (1.0f / 32.0f);
    }
}

extern "C" void kernel_launch(void* const* d_in, const int* in_sizes, int n_in,
                              void* d_out, int out_size, void* d_ws, size_t ws_size,
                              hipStream_t stream) {
    (void)in_sizes; (void)n_in; (void)out_size; (void)ws_size;
    const float*  box_cls  = (const float*) d_in[0];
    const float*  box_reg  = (const float*) d_in[1];
    const float*  iou_pred = (const float*) d_in[2];
    const int4*   pcp      = (const int4*)  d_in[3];
    const int4*   prp      = (const int4*)  d_in[4];
    const float*  plogit   = (const float*) d_in[5];
    const float4* anchors  = (const float4*)d_in[6];
    const float4* tboxes   = (const float4*)d_in[7];
    // d_in[8] = target_labels: unused by the reference outputs
    float* out = (float*)d_out;

    float4* whole = (float4*)d_ws;        // 8*16384*16 B = 2 MB scratch

    decode_whole_kernel<<<(NB * HW) / 256, 256, 0, stream>>>(box_reg, anchors, whole);
    match_kernel<<<NB * TT, 256, 0, stream>>>(whole, tboxes, out, out + 1024);
    peaks_kernel<<<8192 / 256, 256, 0, stream>>>(box_cls, box_reg, iou_pred,
                                                 pcp, prp, plogit, anchors, out);
    mean_kernel<<<1, 32, 0, stream>>>(out);
}